// max_general_2x2_13821204759254
// MI455X (gfx1250) — compile-verified
//
#include <hip/hip_runtime.h>
#include <hip/hip_bf16.h>

// 2x2 max-pool over (32,64,112,112) f32 -> (32,64,56,56) f32.
// Memory-bound: ~128.5 MB total traffic -> ~5.5us floor at 23.3 TB/s.
// Strategy: each block streams 32 contiguous row-pairs (28 KB) into LDS via
// gfx1250 async global->LDS b128 DMA (no VGPR destinations), waits ASYNCcnt,
// then emits 448 float4 outputs with non-temporal b128 stores.

#if defined(__has_builtin)
#  if __has_builtin(__builtin_amdgcn_global_load_async_to_lds_b128)
#    define HAVE_ASYNC_LDS 1
#  endif
#endif
#ifndef HAVE_ASYNC_LDS
#  define HAVE_ASYNC_LDS 0
#endif

typedef float v4f __attribute__((ext_vector_type(4)));
typedef int   v4i __attribute__((ext_vector_type(4)));
typedef __attribute__((address_space(1))) v4i* as1_v4i_ptr;  // global
typedef __attribute__((address_space(3))) v4i* as3_v4i_ptr;  // LDS

#define ROWS_PER_BLOCK 32                       // output rows (input row-pairs) per block
#define CHUNK_FLOATS   (ROWS_PER_BLOCK * 224)   // 7168 floats = 28 KB
#define THREADS        256
#define LOAD_STEPS     (CHUNK_FLOATS / (THREADS * 4))  // 7 async b128 loads/thread
#define TASKS          (ROWS_PER_BLOCK * 14)    // 448 float4 outputs per block

__global__ __launch_bounds__(THREADS)
void maxpool2x2_kernel(const float* __restrict__ in, float* __restrict__ out) {
    const int tid = threadIdx.x;
    const int cid = blockIdx.x;                 // chunk id, exact (114688 rows / 32)
    const float* gbase = in + (long long)cid * CHUNK_FLOATS;

#if HAVE_ASYNC_LDS
    __shared__ float lds[CHUNK_FLOATS];
    #pragma unroll
    for (int i = 0; i < LOAD_STEPS; ++i) {
        const int off = (i * THREADS + tid) * 4;  // 16B per lane, fully coalesced
        __builtin_amdgcn_global_load_async_to_lds_b128(
            (as1_v4i_ptr)(gbase + off),
            (as3_v4i_ptr)(lds + off),
            /*imm offset=*/0, /*cpol=*/0);
    }
#  if __has_builtin(__builtin_amdgcn_s_wait_asynccnt)
    __builtin_amdgcn_s_wait_asynccnt(0);
#  else
    asm volatile("s_wait_asynccnt 0" ::: "memory");
#  endif
    __syncthreads();
#endif

    // 448 tasks: task j -> row-pair k = j/14, group g = j%14 (4 outputs each).
    #pragma unroll
    for (int rep = 0; rep < 2; ++rep) {
        const int j = rep * THREADS + tid;
        if (j < TASKS) {
            const int k = j / 14;
            const int g = j % 14;
#if HAVE_ASYNC_LDS
            // LDS reads (ds_load_b128): row-pair k occupies lds[k*224 .. k*224+223]
            const v4f a0 = *(const v4f*)(lds + k * 224 + g * 8);
            const v4f a1 = *(const v4f*)(lds + k * 224 + g * 8 + 4);
            const v4f b0 = *(const v4f*)(lds + k * 224 + 112 + g * 8);
            const v4f b1 = *(const v4f*)(lds + k * 224 + 112 + g * 8 + 4);
#else
            // Fallback: direct streaming b128 loads with non-temporal hint
            const v4f a0 = __builtin_nontemporal_load((const v4f*)(gbase + k * 224 + g * 8));
            const v4f a1 = __builtin_nontemporal_load((const v4f*)(gbase + k * 224 + g * 8 + 4));
            const v4f b0 = __builtin_nontemporal_load((const v4f*)(gbase + k * 224 + 112 + g * 8));
            const v4f b1 = __builtin_nontemporal_load((const v4f*)(gbase + k * 224 + 112 + g * 8 + 4));
#endif
            v4f o;
            o.x = fmaxf(fmaxf(a0.x, a0.y), fmaxf(b0.x, b0.y));
            o.y = fmaxf(fmaxf(a0.z, a0.w), fmaxf(b0.z, b0.w));
            o.z = fmaxf(fmaxf(a1.x, a1.y), fmaxf(b1.x, b1.y));
            o.w = fmaxf(fmaxf(a1.z, a1.w), fmaxf(b1.z, b1.w));

            const long long R = (long long)cid * ROWS_PER_BLOCK + k;  // global output row
            __builtin_nontemporal_store(o, (v4f*)(out + R * 56 + g * 4));
        }
    }
}

extern "C" void kernel_launch(void* const* d_in, const int* in_sizes, int n_in,
                              void* d_out, int out_size, void* d_ws, size_t ws_size,
                              hipStream_t stream) {
    const float* x = (const float*)d_in[0];
    float* out = (float*)d_out;

    // rows = N*C*Hout = in_elems / 224; divides ROWS_PER_BLOCK exactly for this shape.
    const int rows   = in_sizes[0] / 224;       // 114688
    const int blocks = rows / ROWS_PER_BLOCK;   // 3584

    maxpool2x2_kernel<<<blocks, THREADS, 0, stream>>>(x, out);
}